// RNN_9835475108470
// MI455X (gfx1250) — compile-verified
//
#include <hip/hip_runtime.h>
#include <math.h>

// ---------------------------------------------------------------------------
// Problem dims (match reference)
// ---------------------------------------------------------------------------
constexpr int Bc = 64, Tc = 512, Vc = 4096, Ec = 512, Hc = 512;
constexpr int BT = Bc * Tc;                       // 32768 rows
constexpr size_t BUF_ELTS = (size_t)BT * Hc;      // xproj/hidden buffer (in-place)

typedef __attribute__((ext_vector_type(2))) float v2f;
typedef __attribute__((ext_vector_type(8))) float v8f;

// D = A(16x4) * B(4x16) + C, f32, wave32.
// 8-arg form: (neg_a, A, neg_b, B, c_mod, C, reuse_a, reuse_b)
static __device__ __forceinline__ v8f wmma_f32(v2f a, v2f b, v8f c) {
  return __builtin_amdgcn_wmma_f32_16x16x4_f32(false, a, false, b, (short)0, c,
                                               false, false);
}

// ---------------------------------------------------------------------------
// Kernel 1: xproj[bt][h] = wte[idx[bt]][:] @ Wx + b_xh   (gather-GEMM)
// wave tile 32x64 (2x4 WMMA tiles), block 256 thr = 8 waves -> 64x256 tile
// ---------------------------------------------------------------------------
__global__ __launch_bounds__(256)
void xproj_kernel(const int* __restrict__ idx, const float* __restrict__ wte,
                  const float* __restrict__ W_xh, const float* __restrict__ b_xh,
                  float* __restrict__ buf) {
  const int lane = threadIdx.x & 31;
  const int w    = threadIdx.x >> 5;
  const int wm = w >> 2, wn = w & 3;
  const int l16 = lane & 15, hk = lane >> 4;
  const int m0 = blockIdx.x * 64 + wm * 32;
  const int n0 = blockIdx.y * 256 + wn * 64;

  // gathered A rows (token embeddings), contiguous in K
  const float* a0p = wte + (size_t)idx[m0 + l16] * Ec;
  const float* a1p = wte + (size_t)idx[m0 + 16 + l16] * Ec;

  v8f acc[2][4];
#pragma unroll
  for (int i = 0; i < 2; ++i)
#pragma unroll
    for (int j = 0; j < 4; ++j) acc[i][j] = {};

#pragma unroll 4
  for (int k0 = 0; k0 < Ec; k0 += 4) {
    const int kb = k0 + 2 * hk;
    v2f a0 = *(const v2f*)(a0p + kb);
    v2f a1 = *(const v2f*)(a1p + kb);
#pragma unroll
    for (int j = 0; j < 4; ++j) {
      const int col = n0 + j * 16 + l16;
      v2f b;
      b.x = W_xh[(size_t)kb * Hc + col];
      b.y = W_xh[(size_t)(kb + 1) * Hc + col];
      acc[0][j] = wmma_f32(a0, b, acc[0][j]);
      acc[1][j] = wmma_f32(a1, b, acc[1][j]);
    }
  }

#pragma unroll
  for (int j = 0; j < 4; ++j) {
    const int col  = n0 + j * 16 + l16;
    const float bv = b_xh[col];
#pragma unroll
    for (int i = 0; i < 2; ++i)
#pragma unroll
      for (int e = 0; e < 8; ++e) {
        const int row = m0 + i * 16 + 8 * hk + e;
        buf[(size_t)row * Hc + col] = acc[i][j][e] + bv;
      }
  }
}

// ---------------------------------------------------------------------------
// Kernel 2: recurrence  h_t = tanh(xproj_t + h_{t-1} @ Wh), in-place in buf.
// 4 independent batch groups (16 rows) x 8 col-blocks; per-step release/acquire
// spin barrier among each group's 8 blocks.  128 thr = 4 waves, 16 cols/wave.
// ---------------------------------------------------------------------------
__global__ __launch_bounds__(128)
void rnn_kernel(const float* __restrict__ start, const float* __restrict__ W_xh,
                float* __restrict__ buf, unsigned* __restrict__ bar) {
  const int lane = threadIdx.x & 31;
  const int w    = threadIdx.x >> 5;
  const int l16 = lane & 15, hk = lane >> 4;
  const int g  = blockIdx.x >> 3;          // batch group 0..3
  const int nb = blockIdx.x & 7;           // col block 0..7
  const int mbase = g * 16;
  const int col = nb * 64 + w * 16 + l16;
  const int brow = mbase + l16;            // this lane's A (batch) row
  const float* Wh = W_xh + (size_t)Ec * Hc;

  for (int t = 0; t < Tc; ++t) {
    v8f acc = {};
    if (t == 0) {
      // h0 = start row broadcast to all batch rows
#pragma unroll 8
      for (int k0 = 0; k0 < Hc; k0 += 4) {
        const int kb = k0 + 2 * hk;
        v2f a = *(const v2f*)(start + kb);
        v2f b;
        b.x = Wh[(size_t)kb * Hc + col];
        b.y = Wh[(size_t)(kb + 1) * Hc + col];
        acc = wmma_f32(a, b, acc);
      }
    } else {
      const float* hp = buf + ((size_t)brow * Tc + (t - 1)) * Hc;
#pragma unroll 8
      for (int k0 = 0; k0 < Hc; k0 += 4) {
        const int kb = k0 + 2 * hk;
        v2f a = *(const v2f*)(hp + kb);
        v2f b;
        b.x = Wh[(size_t)kb * Hc + col];
        b.y = Wh[(size_t)(kb + 1) * Hc + col];
        acc = wmma_f32(a, b, acc);
      }
    }
    // epilogue: h_new = tanh(xproj + acc), stored over xproj slot (in-place)
#pragma unroll
    for (int e = 0; e < 8; ++e) {
      const int r = mbase + 8 * hk + e;
      const size_t o = ((size_t)r * Tc + t) * Hc + col;
      buf[o] = tanhf(acc[e] + buf[o]);
    }
    // group barrier: all 8 blocks of this group finished step t
    __threadfence();
    __syncthreads();
    if (threadIdx.x == 0) {
      __hip_atomic_fetch_add(bar + g, 1u, __ATOMIC_RELEASE,
                             __HIP_MEMORY_SCOPE_AGENT);
      const unsigned target = 8u * (unsigned)(t + 1);
      while (__hip_atomic_load(bar + g, __ATOMIC_ACQUIRE,
                               __HIP_MEMORY_SCOPE_AGENT) < target)
        __builtin_amdgcn_s_sleep(2);
    }
    __syncthreads();
  }
}

// ---------------------------------------------------------------------------
// Kernel 3: logits = hidden @ W_head + b_head  (dominant GEMM, 137 GFLOP)
// same 32x64-per-wave tiling; W_head (8 MB) is L2-resident.
// ---------------------------------------------------------------------------
__global__ __launch_bounds__(256)
void head_kernel(const float* __restrict__ buf, const float* __restrict__ W_head,
                 const float* __restrict__ b_head, float* __restrict__ out) {
  const int lane = threadIdx.x & 31;
  const int w    = threadIdx.x >> 5;
  const int wm = w >> 2, wn = w & 3;
  const int l16 = lane & 15, hk = lane >> 4;
  const int m0 = blockIdx.x * 64 + wm * 32;
  const int n0 = blockIdx.y * 256 + wn * 64;

  const float* a0p = buf + (size_t)(m0 + l16) * Hc;
  const float* a1p = buf + (size_t)(m0 + 16 + l16) * Hc;

  v8f acc[2][4];
#pragma unroll
  for (int i = 0; i < 2; ++i)
#pragma unroll
    for (int j = 0; j < 4; ++j) acc[i][j] = {};

#pragma unroll 4
  for (int k0 = 0; k0 < Hc; k0 += 4) {
    const int kb = k0 + 2 * hk;
    v2f a0 = *(const v2f*)(a0p + kb);
    v2f a1 = *(const v2f*)(a1p + kb);
#pragma unroll
    for (int j = 0; j < 4; ++j) {
      const int col = n0 + j * 16 + l16;
      v2f b;
      b.x = W_head[(size_t)kb * Vc + col];
      b.y = W_head[(size_t)(kb + 1) * Vc + col];
      acc[0][j] = wmma_f32(a0, b, acc[0][j]);
      acc[1][j] = wmma_f32(a1, b, acc[1][j]);
    }
  }

#pragma unroll
  for (int j = 0; j < 4; ++j) {
    const int col  = n0 + j * 16 + l16;
    const float bv = b_head[col];
#pragma unroll
    for (int i = 0; i < 2; ++i)
#pragma unroll
      for (int e = 0; e < 8; ++e) {
        const int row = m0 + i * 16 + 8 * hk + e;
        out[(size_t)row * Vc + col] = acc[i][j][e] + bv;
      }
  }
}

// ---------------------------------------------------------------------------
extern "C" void kernel_launch(void* const* d_in, const int* in_sizes, int n_in,
                              void* d_out, int out_size, void* d_ws,
                              size_t ws_size, hipStream_t stream) {
  (void)in_sizes; (void)n_in; (void)out_size; (void)ws_size;
  const int*   idx    = (const int*)d_in[0];
  const float* start  = (const float*)d_in[1];
  const float* wte    = (const float*)d_in[2];
  const float* W_xh   = (const float*)d_in[3];
  const float* b_xh   = (const float*)d_in[4];
  const float* W_head = (const float*)d_in[5];
  const float* b_head = (const float*)d_in[6];
  float* out = (float*)d_out;

  float*    buf = (float*)d_ws;                                  // 64 MB
  unsigned* bar = (unsigned*)((char*)d_ws + BUF_ELTS * sizeof(float));

  hipMemsetAsync(bar, 0, 4 * sizeof(unsigned), stream);          // reset barrier

  xproj_kernel<<<dim3(BT / 64, Hc / 256), 256, 0, stream>>>(idx, wte, W_xh,
                                                            b_xh, buf);
  rnn_kernel<<<32, 128, 0, stream>>>(start, W_xh, buf, bar);
  head_kernel<<<dim3(BT / 64, Vc / 256), 256, 0, stream>>>(buf, W_head, b_head,
                                                           out);
}